// Self_Attention_15702400434565
// MI455X (gfx1250) — compile-verified
//
#include <hip/hip_runtime.h>
#include <math.h>
#include <stdint.h>

// Problem constants (match reference)
enum { NB = 4, NS = 1024, NHID = 512, NH = 8, ND = 64, NNB = 32 };

typedef float v2f __attribute__((ext_vector_type(2)));
typedef float v8f __attribute__((ext_vector_type(8)));

// ---------------------------------------------------------------------------
// Kernel 1: per (b, h, 16-query tile):
//   top-32 selection on attention_weight rows (async global->LDS double
//   buffered stream), QK^T via v_wmma_f32_16x16x4_f32 with online softmax,
//   sparse renormalized PV, write pre-LayerNorm output.
// ---------------------------------------------------------------------------
__global__ __launch_bounds__(128) void attn_topk_kernel(
    const float* __restrict__ qg, const float* __restrict__ kg,
    const float* __restrict__ vg, const float* __restrict__ aw,
    float* __restrict__ pre_out)
{
    __shared__ float sQ[16 * 64];              // Q tile
    __shared__ int   sSlotI[16 * 1024 / 4];    // byte slot table (int-aliased)
    __shared__ int   sIdx[16 * 32];            // selected column per (row, slot)
    __shared__ float sSelBuf[16 * 33];         // raw scores (+1 dummy slot/row)
    __shared__ float sPm[4 * 16], sPz[4 * 16];
    __shared__ float sMax[16], sInv[16];
    __shared__ float sAW[4 * 2 * 1024];        // per-wave double buffer (4KB x2)

    unsigned char* sSlot = reinterpret_cast<unsigned char*>(sSlotI);

    const int tid  = threadIdx.x;
    const int wave = tid >> 5;
    const int lane = tid & 31;

    const int bid = blockIdx.x;        // 0 .. NB*NH*64-1
    const int b   = bid >> 9;          // NH*64 = 512 tiles per batch
    const int h   = (bid >> 6) & 7;
    const int qt  = bid & 63;
    const int q0  = qt * 16;

    // ---- Phase A: stage Q tile, clear slot table ----
    {
        const float* qbase = qg + ((size_t)(b * NS + q0)) * NHID + h * ND;
        for (int i = tid; i < 16 * 64; i += 128) {
            int r = i >> 6, c = i & 63;
            sQ[i] = qbase[(size_t)r * NHID + c];
        }
        for (int i = tid; i < 16 * 1024 / 4; i += 128) sSlotI[i] = -1;
    }
    __syncthreads();

    // ---- Phase B: exact top-32 per row (ties -> smaller column index) ----
    // attention_weight rows stream through LDS via async copies, double
    // buffered so row r+1's copy overlaps row r's 32-round selection.
    {
        float* mybuf = &sAW[wave * 2 * 1024];

        auto awbase = [&](int rr) -> const float* {
            return aw + ((size_t)((b * NH + h) * NS + (q0 + wave * 4 + rr))) * NS;
        };
        auto issue_async_row = [&](const float* gsrc, int dstbuf) {
            unsigned ldsbase = (unsigned)(uintptr_t)(mybuf + dstbuf * 1024);
            #pragma unroll
            for (int i = 0; i < 8; ++i) {
                unsigned     loff = ldsbase + (unsigned)(i * 512 + lane * 16);
                const float* ga   = gsrc + i * 128 + lane * 4;
                asm volatile("global_load_async_to_lds_b128 %0, %1, off"
                             :: "v"(loff), "v"(ga) : "memory");
            }
        };

        issue_async_row(awbase(0), 0);           // preload row 0

        for (int rr = 0; rr < 4; ++rr) {
            const int row = wave * 4 + rr;
            const int cur = rr & 1;

            // current buffer's copies complete
            asm volatile("s_wait_asynccnt 0x0" ::: "memory");

            if (rr < 3) {
                issue_async_row(awbase(rr + 1), 1 - cur);   // overlap next row
            }
            if (rr < 2) {
                __builtin_prefetch(awbase(rr + 2) + lane * 32, 0, 0); // warm L2
            }

            float wv[32];
            #pragma unroll
            for (int j = 0; j < 32; ++j) wv[j] = mybuf[cur * 1024 + j * 32 + lane];
            unsigned selmask = 0u;

            for (int it = 0; it < NNB; ++it) {
                float bestv = -3.4e38f;
                int   bestc = 1 << 30;
                #pragma unroll
                for (int j = 0; j < 32; ++j) {
                    if (!((selmask >> j) & 1u)) {
                        float x = wv[j];
                        int   c = j * 32 + lane;
                        if (x > bestv || (x == bestv && c < bestc)) { bestv = x; bestc = c; }
                    }
                }
                #pragma unroll
                for (int off = 16; off >= 1; off >>= 1) {
                    float ov = __shfl_xor(bestv, off, 32);
                    int   oc = __shfl_xor(bestc, off, 32);
                    if (ov > bestv || (ov == bestv && oc < bestc)) { bestv = ov; bestc = oc; }
                }
                if ((bestc & 31) == lane) selmask |= 1u << (bestc >> 5);
                if (lane == 0) {
                    sIdx[row * 32 + it]       = bestc;
                    sSlot[row * 1024 + bestc] = (unsigned char)it;
                }
            }
        }
    }
    __syncthreads();

    // ---- Phase C: QK^T via WMMA f32 16x16x4, online softmax in registers ----
    {
        const int hi = lane >> 4;     // half-wave select (K sub-pair)
        const int ml = lane & 15;

        // A fragments: 16 chunks of K-dim (each 16x4): lane ml holds row ml
        v2f afrag[16];
        #pragma unroll
        for (int kk = 0; kk < 16; ++kk)
            afrag[kk] = *(const v2f*)(&sQ[ml * 64 + kk * 4 + hi * 2]);

        float m8[8], z8[8];
        #pragma unroll
        for (int r = 0; r < 8; ++r) { m8[r] = -3.4e38f; z8[r] = 0.0f; }

        for (int t = 0; t < 16; ++t) {
            const int nt  = wave * 16 + t;      // column tile
            const int n   = nt * 16 + ml;       // key row this lane supplies
            const float* kp =
                kg + ((size_t)(b * NS + n)) * NHID + h * ND + hi * 2;

            v8f acc = {};
            #pragma unroll
            for (int kk = 0; kk < 16; ++kk) {
                v2f bfrag = *(const v2f*)(kp + kk * 4);
                acc = __builtin_amdgcn_wmma_f32_16x16x4_f32(
                    false, afrag[kk], false, bfrag, (short)0, acc, false, false);
            }

            const int col = nt * 16 + ml;
            #pragma unroll
            for (int r = 0; r < 8; ++r) {
                float sc  = acc[r];
                int   row = r + hi * 8;
                // branch-free scatter: unselected -> dummy slot 32
                int slot = (int)sSlot[row * 1024 + col];
                slot = (slot > 31) ? 32 : slot;
                sSelBuf[row * 33 + slot] = sc;            // raw score
                // online softmax
                float mo = m8[r];
                if (sc > mo) { z8[r] *= __expf(mo - sc); m8[r] = sc; mo = sc; }
                z8[r] += __expf(sc - mo);
            }
        }

        // merge (m, z) across the 16 lanes of each half-wave
        #pragma unroll
        for (int r = 0; r < 8; ++r) {
            float m = m8[r], z = z8[r];
            #pragma unroll
            for (int off = 1; off < 16; off <<= 1) {
                float om = __shfl_xor(m, off, 32);
                float oz = __shfl_xor(z, off, 32);
                float nm = fmaxf(m, om);
                z = z * __expf(m - nm) + oz * __expf(om - nm);
                m = nm;
            }
            m8[r] = m; z8[r] = z;
        }
        if (ml == 0) {
            #pragma unroll
            for (int r = 0; r < 8; ++r) {
                int row = r + hi * 8;
                sPm[wave * 16 + row] = m8[r];
                sPz[wave * 16 + row] = z8[r];
            }
        }
    }
    __syncthreads();

    // ---- Phase D: merge waves, renormalization factor per row ----
    if (tid < 16) {
        float m = sPm[tid], z = sPz[tid];
        for (int w = 1; w < 4; ++w) {
            float om = sPm[w * 16 + tid], oz = sPz[w * 16 + tid];
            float nm = fmaxf(m, om);
            z = z * __expf(m - nm) + oz * __expf(om - nm);
            m = nm;
        }
        float ts = 0.0f;
        for (int j = 0; j < 32; ++j) ts += __expf(sSelBuf[tid * 33 + j] - m);
        sMax[tid] = m;
        sInv[tid] = 1.0f / (ts + 1e-5f * z);   // folds softmax Z + eps renorm
    }
    __syncthreads();

    // raw selected scores -> final renormalized weights (in place)
    for (int i = tid; i < 16 * 32; i += 128) {
        int row = i >> 5, j = i & 31;
        sSelBuf[row * 33 + j] =
            __expf(sSelBuf[row * 33 + j] - sMax[row]) * sInv[row];
    }
    __syncthreads();

    // ---- Phase E: sparse PV gather (32 nonzeros per row), write pre-LN ----
    {
        const int row = tid >> 3;
        const int db  = (tid & 7) * 8;
        float o0=0,o1=0,o2=0,o3=0,o4=0,o5=0,o6=0,o7=0;
        for (int j = 0; j < 32; ++j) {
            int   col = sIdx[row * 32 + j];
            float wj  = sSelBuf[row * 33 + j];
            const float* vp =
                vg + ((size_t)(b * NS + col)) * NHID + h * ND + db;
            float4 p0 = *(const float4*)(vp);
            float4 p1 = *(const float4*)(vp + 4);
            o0 += wj * p0.x; o1 += wj * p0.y; o2 += wj * p0.z; o3 += wj * p0.w;
            o4 += wj * p1.x; o5 += wj * p1.y; o6 += wj * p1.z; o7 += wj * p1.w;
        }
        float* op = pre_out + ((size_t)(b * NS + q0 + row)) * NHID + h * ND + db;
        float4 w0 = {o0, o1, o2, o3}, w1 = {o4, o5, o6, o7};
        *(float4*)(op)     = w0;
        *(float4*)(op + 4) = w1;
    }
}

// ---------------------------------------------------------------------------
// Kernel 2: LayerNorm over last dim (512) per row of [B*S].
// ---------------------------------------------------------------------------
__global__ __launch_bounds__(128) void layernorm_kernel(
    const float* __restrict__ xin, const float* __restrict__ gamma,
    const float* __restrict__ beta, float* __restrict__ out)
{
    __shared__ float sSum[4], sSq[4];
    const int row = blockIdx.x;
    const int tid = threadIdx.x;
    const float* x = xin + (size_t)row * NHID;

    float4 a = *(const float4*)(x + tid * 4);
    float s  = a.x + a.y + a.z + a.w;
    float sq = a.x * a.x + a.y * a.y + a.z * a.z + a.w * a.w;
    #pragma unroll
    for (int off = 16; off >= 1; off >>= 1) {
        s  += __shfl_xor(s,  off, 32);
        sq += __shfl_xor(sq, off, 32);
    }
    if ((tid & 31) == 0) { sSum[tid >> 5] = s; sSq[tid >> 5] = sq; }
    __syncthreads();

    float ts = sSum[0] + sSum[1] + sSum[2] + sSum[3];
    float tq = sSq[0]  + sSq[1]  + sSq[2]  + sSq[3];
    float mu  = ts * (1.0f / NHID);
    float var = tq * (1.0f / NHID) - mu * mu;
    float sc  = rsqrtf(var + 1e-5f);

    float4 g  = *(const float4*)(gamma + tid * 4);
    float4 be = *(const float4*)(beta  + tid * 4);
    float4 r;
    r.x = (a.x - mu) * sc * g.x + be.x;
    r.y = (a.y - mu) * sc * g.y + be.y;
    r.z = (a.z - mu) * sc * g.z + be.z;
    r.w = (a.w - mu) * sc * g.w + be.w;
    *(float4*)(out + (size_t)row * NHID + tid * 4) = r;
}

// ---------------------------------------------------------------------------
extern "C" void kernel_launch(void* const* d_in, const int* in_sizes, int n_in,
                              void* d_out, int out_size, void* d_ws, size_t ws_size,
                              hipStream_t stream) {
    const float* q     = (const float*)d_in[0];
    const float* k     = (const float*)d_in[1];
    const float* v     = (const float*)d_in[2];
    const float* aw    = (const float*)d_in[3];
    const float* gamma = (const float*)d_in[4];
    const float* beta  = (const float*)d_in[5];
    float* pre = (float*)d_ws;   // [B,S,HID] pre-LayerNorm scratch (8 MB)

    attn_topk_kernel<<<NB * NH * (NS / 16), 128, 0, stream>>>(q, k, v, aw, pre);
    layernorm_kernel<<<NB * NS, 128, 0, stream>>>(pre, gamma, beta, (float*)d_out);
}